// rails2track_11914239279247
// MI455X (gfx1250) — compile-verified
//
#include <hip/hip_runtime.h>
#include <hip/hip_bf16.h>

#define H 4096
#define W 4096
#define RAILS 1

typedef __attribute__((ext_vector_type(2))) float v2f;
typedef __attribute__((ext_vector_type(8))) float v8f;

// ---------------------------------------------------------------------------
// Kernel 0: zero the 16x16 Gram/moment accumulator
// ---------------------------------------------------------------------------
__global__ void zero_gram_kernel(float* __restrict__ gram) {
    if (threadIdx.x < 256) gram[threadIdx.x] = 0.0f;
}

// ---------------------------------------------------------------------------
// Kernel 1: per-row stats — transition count, first rail col, last rail col
// One block (256 threads) per row; B128 loads; LDS tree reduction.
// ---------------------------------------------------------------------------
__global__ void row_stats_kernel(const int* __restrict__ mask,
                                 int* __restrict__ first,
                                 int* __restrict__ last,
                                 int* __restrict__ clusters) {
    const int row = blockIdx.x;
    const int tid = threadIdx.x;
    const int* mrow = mask + (size_t)row * W;

    int trans = 0;
    int mn = W;       // min col with rail
    int mx = -1;      // max col with rail

    // 4096 cols / (256 thr * 4 per int4) = 4 chunks per thread
    #pragma unroll
    for (int k = 0; k < 4; ++k) {
        const int c0 = 4 * (tid + 256 * k);
        const int4 a = *(const int4*)(mrow + c0);
        const int prev = (c0 == 0) ? a.x : mrow[c0 - 1];
        trans += (a.x != prev) + (a.y != a.x) + (a.z != a.y) + (a.w != a.z);
        if (a.x == RAILS) { mn = min(mn, c0 + 0); mx = max(mx, c0 + 0); }
        if (a.y == RAILS) { mn = min(mn, c0 + 1); mx = max(mx, c0 + 1); }
        if (a.z == RAILS) { mn = min(mn, c0 + 2); mx = max(mx, c0 + 2); }
        if (a.w == RAILS) { mn = min(mn, c0 + 3); mx = max(mx, c0 + 3); }
    }

    __shared__ int s_t[256], s_mn[256], s_mx[256];
    s_t[tid] = trans; s_mn[tid] = mn; s_mx[tid] = mx;
    __syncthreads();
    for (int off = 128; off > 0; off >>= 1) {
        if (tid < off) {
            s_t[tid]  += s_t[tid + off];
            s_mn[tid]  = min(s_mn[tid], s_mn[tid + off]);
            s_mx[tid]  = max(s_mx[tid], s_mx[tid + off]);
        }
        __syncthreads();
    }
    if (tid == 0) {
        const bool any = (s_mx[0] >= 0);
        clusters[row] = s_t[0] >> 1;            // transitions / 2
        first[row]    = any ? s_mn[0] : 0;      // argmax semantics on all-zero row
        last[row]     = any ? (s_mx[0] + 1) : W; // W - argmax(reversed)
    }
}

// ---------------------------------------------------------------------------
// Kernel 2: WMMA tall-skinny reduction  D = X^T * [X | yF | yL]
//   A (16x4)  = X^T chunk      A[i][k] = s_r * xn_r^i   (i<=8, else 0)
//   B (4x16)  = [X | y] chunk  B[k][j] = s_r * xn_r^j (j<=8), s*first/1024 (j==9),
//                              s*last/1024 (j==10), 0 otherwise
// Per ISA VGPR layouts (wave32):
//   A: M = lane&15; VGPR0/1 hold K = 2*(lane>>4) / +1
//   B: N = lane&15; VGPR0/1 hold K = 2*(lane>>4) / +1
//   D: element e -> M = e + 8*(lane>>4), N = lane&15
// ---------------------------------------------------------------------------
__global__ void gram_wmma_kernel(const int* __restrict__ first,
                                 const int* __restrict__ last,
                                 const int* __restrict__ clusters,
                                 float* __restrict__ gram) {
    const int lane   = threadIdx.x & 31;
    const int wave   = blockIdx.x * (blockDim.x >> 5) + (threadIdx.x >> 5);
    const int nwaves = gridDim.x * (blockDim.x >> 5);
    const int half   = lane >> 4;
    const int idx    = lane & 15;

    v8f acc = {};

    for (int r0 = wave * 4; r0 < H; r0 += nwaves * 4) {
        const int ra = r0 + 2 * half;
        const int rb = ra + 1;

        const float sa = (clusters[ra] == 2) ? 1.0f : 0.0f;
        const float sb = (clusters[rb] == 2) ? 1.0f : 0.0f;
        const float xa = (float)ra * (1.0f / 1024.0f);
        const float xb = (float)rb * (1.0f / 1024.0f);

        // branchless xn^idx via square-and-multiply selects (keeps EXEC uniform)
        float pa = 1.0f, pb = 1.0f;
        {
            float a1 = xa, b1 = xb;
            float a2 = a1 * a1, b2 = b1 * b1;
            float a4 = a2 * a2, b4 = b2 * b2;
            float a8 = a4 * a4, b8 = b4 * b4;
            pa *= (idx & 1) ? a1 : 1.0f;  pb *= (idx & 1) ? b1 : 1.0f;
            pa *= (idx & 2) ? a2 : 1.0f;  pb *= (idx & 2) ? b2 : 1.0f;
            pa *= (idx & 4) ? a4 : 1.0f;  pb *= (idx & 4) ? b4 : 1.0f;
            pa *= (idx & 8) ? a8 : 1.0f;  pb *= (idx & 8) ? b8 : 1.0f;
        }

        v2f A, B;
        A.x = (idx <= 8) ? sa * pa : 0.0f;
        A.y = (idx <= 8) ? sb * pb : 0.0f;

        float ca, cb;
        if (idx <= 8)       { ca = sa * pa;                                cb = sb * pb; }
        else if (idx == 9)  { ca = sa * (float)first[ra] * (1.0f/1024.0f); cb = sb * (float)first[rb] * (1.0f/1024.0f); }
        else if (idx == 10) { ca = sa * (float)last[ra]  * (1.0f/1024.0f); cb = sb * (float)last[rb]  * (1.0f/1024.0f); }
        else                { ca = 0.0f;                                   cb = 0.0f; }
        B.x = ca;
        B.y = cb;

        acc = __builtin_amdgcn_wmma_f32_16x16x4_f32(
            /*neg_a=*/false, A, /*neg_b=*/false, B,
            /*c_mod=*/(short)0, acc, /*reuse_a=*/false, /*reuse_b=*/false);
    }

    #pragma unroll
    for (int e = 0; e < 8; ++e) {
        const int M = e + 8 * half;
        atomicAdd(&gram[M * 16 + idx], acc[e]);
    }
}

// ---------------------------------------------------------------------------
// Kernel 3: solve the 9x9 / 3x3 normal equations (double-precision Gaussian
// elimination with partial pivoting; equivalent to the SVD pinv for this
// full-rank Vandermonde design).
// coef layout: cL1[0..8] cL2[9..11] cR1[12..20] cR2[21..23]
//              cC1[24..32] cC2[33..35] cD[36..38]
// ---------------------------------------------------------------------------
__device__ void solveN(const double* G, const double* b, int n, float* out) {
    double A[9][10];
    double x[9];
    for (int i = 0; i < n; ++i) {
        for (int j = 0; j < n; ++j) A[i][j] = G[i * 9 + j];
        A[i][n] = b[i];
    }
    for (int k = 0; k < n; ++k) {
        int p = k; double best = fabs(A[k][k]);
        for (int i = k + 1; i < n; ++i) {
            double v = fabs(A[i][k]);
            if (v > best) { best = v; p = i; }
        }
        if (p != k) for (int j = k; j <= n; ++j) { double t = A[k][j]; A[k][j] = A[p][j]; A[p][j] = t; }
        double piv = A[k][k];
        if (piv == 0.0) piv = 1e-300;
        double inv = 1.0 / piv;
        for (int i = k + 1; i < n; ++i) {
            double f = A[i][k] * inv;
            for (int j = k; j <= n; ++j) A[i][j] -= f * A[k][j];
        }
    }
    for (int i = n - 1; i >= 0; --i) {
        double s = A[i][n];
        for (int j = i + 1; j < n; ++j) s -= A[i][j] * x[j];
        x[i] = s / A[i][i];
    }
    for (int i = 0; i < n; ++i) out[i] = (float)x[i];
}

__global__ void solve_fits_kernel(const float* __restrict__ gram,
                                  float* __restrict__ coef) {
    if (threadIdx.x != 0 || blockIdx.x != 0) return;
    double G[81], bF[9], bL[9], bM[9], bD[9];
    for (int i = 0; i < 9; ++i) {
        for (int j = 0; j < 9; ++j) G[i * 9 + j] = (double)gram[i * 16 + j];
        bF[i] = (double)gram[i * 16 + 9];
        bL[i] = (double)gram[i * 16 + 10];
        bM[i] = 0.5 * (bF[i] + bL[i]);
        bD[i] = bL[i] - bF[i];
    }
    solveN(G, bF, 9, coef + 0);   // cL1
    solveN(G, bF, 3, coef + 9);   // cL2
    solveN(G, bL, 9, coef + 12);  // cR1
    solveN(G, bL, 3, coef + 21);  // cR2
    solveN(G, bM, 9, coef + 24);  // cC1
    solveN(G, bM, 3, coef + 33);  // cC2
    solveN(G, bD, 3, coef + 36);  // cD
}

// ---------------------------------------------------------------------------
// Kernel 4: per-row border fill + relabel. One block per row; every thread
// redundantly evaluates the scalar border math (keeps EXEC uniform & cheap),
// then the block streams the row with B128 traffic (read should hit L2:
// 64 MB mask << 192 MB L2).
// Final value:  filled&&orig==1 -> 1 ; filled&&orig==0 -> 2 ; else 0
// ---------------------------------------------------------------------------
__global__ void fill_out_kernel(const int* __restrict__ mask,
                                const int* __restrict__ first,
                                const int* __restrict__ last,
                                const int* __restrict__ clusters,
                                const float* __restrict__ coef,
                                float* __restrict__ out) {
    const int row = blockIdx.x;
    const float xn = (float)row * (1.0f / 1024.0f);

    // Horner evaluations
    float h9L = coef[8],  h9R = coef[12 + 8], h9C = coef[24 + 8];
    #pragma unroll
    for (int i = 7; i >= 0; --i) {
        h9L = h9L * xn + coef[i];
        h9R = h9R * xn + coef[12 + i];
        h9C = h9C * xn + coef[24 + i];
    }
    const float h3L = (coef[11] * xn + coef[10]) * xn + coef[9];
    const float h3R = (coef[23] * xn + coef[22]) * xn + coef[21];
    const float h3C = (coef[35] * xn + coef[34]) * xn + coef[33];
    const float h3D = (coef[38] * xn + coef[37]) * xn + coef[36];

    const bool lowrow = ((float)row < 256.0f);
    const float pL = (lowrow ? h9L : h3L) * 1024.0f;
    const float pR = (lowrow ? h9R : h3R) * 1024.0f;
    const float pC = (lowrow ? h9C : h3C) * 1024.0f;
    const float deltas = fabsf(h3D * 1024.0f);   // predict(cD,cD): both branches deg-2

    const int cl = clusters[row];
    const int fi = first[row];
    const int la = last[row];

    int lB = 0, rB = -1;
    bool filled = false;
    if (cl == 2) {
        const float d = (float)(la - fi);
        const int l = (int)((float)fi - 0.2f * d);   // cast (trunc) then clip
        const int r = (int)((float)la + 0.2f * d);
        lB = min(max(l, 0), W - 1);
        rB = min(max(r, 0), W - 1);
        filled = true;
    } else if (cl == 1) {
        const float idxf = (float)fi;                 // argmax == first for 1-cluster rows
        const bool cond = (pC > idxf);
        float lf = cond ? (idxf - 0.2f * deltas) : (idxf - 1.2f * deltas);
        float rf = cond ? (idxf + 1.2f * deltas) : (idxf + 0.2f * deltas);
        lf = fminf(fmaxf(lf, 0.0f), (float)(W - 1));  // clip then cast
        rf = fminf(fmaxf(rf, 0.0f), (float)(W - 1));
        lB = (int)lf; rB = (int)rf;
        filled = true;
    } else if (cl > 2) {
        const float center = 0.5f * (pL + pR);
        float lf = fminf(fmaxf(center - 0.7f * deltas, 0.0f), (float)(W - 1));
        float rf = fminf(fmaxf(center + 0.7f * deltas, 0.0f), (float)(W - 1));
        lB = (int)lf; rB = (int)rf;
        filled = true;
    }

    const int* mrow = mask + (size_t)row * W;
    float* orow = out + (size_t)row * W;

    #pragma unroll
    for (int k = 0; k < 4; ++k) {
        const int c0 = 4 * (threadIdx.x + 256 * k);
        const int4 m = *(const int4*)(mrow + c0);
        float4 o;
        const bool i0 = filled && (c0 + 0 >= lB) && (c0 + 0 <= rB);
        const bool i1 = filled && (c0 + 1 >= lB) && (c0 + 1 <= rB);
        const bool i2 = filled && (c0 + 2 >= lB) && (c0 + 2 <= rB);
        const bool i3 = filled && (c0 + 3 >= lB) && (c0 + 3 <= rB);
        o.x = i0 ? ((m.x == RAILS) ? 1.0f : 2.0f) : 0.0f;
        o.y = i1 ? ((m.y == RAILS) ? 1.0f : 2.0f) : 0.0f;
        o.z = i2 ? ((m.z == RAILS) ? 1.0f : 2.0f) : 0.0f;
        o.w = i3 ? ((m.w == RAILS) ? 1.0f : 2.0f) : 0.0f;
        *(float4*)(orow + c0) = o;
    }
}

// ---------------------------------------------------------------------------
// Launch
// ---------------------------------------------------------------------------
extern "C" void kernel_launch(void* const* d_in, const int* in_sizes, int n_in,
                              void* d_out, int out_size, void* d_ws, size_t ws_size,
                              hipStream_t stream) {
    const int* mask = (const int*)d_in[0];
    float* out = (float*)d_out;

    // workspace carve-up
    char* ws = (char*)d_ws;
    int*   first    = (int*)(ws);                  // 4096 ints
    int*   last     = (int*)(ws + 16 * 1024);      // 4096 ints
    int*   clusters = (int*)(ws + 32 * 1024);      // 4096 ints
    float* gram     = (float*)(ws + 48 * 1024);    // 256 floats (16x16)
    float* coef     = (float*)(ws + 52 * 1024);    // 39 floats

    zero_gram_kernel<<<1, 256, 0, stream>>>(gram);
    row_stats_kernel<<<H, 256, 0, stream>>>(mask, first, last, clusters);
    gram_wmma_kernel<<<16, 256, 0, stream>>>(first, last, clusters, gram); // 128 waves, 8 K-chunks each
    solve_fits_kernel<<<1, 32, 0, stream>>>(gram, coef);
    fill_out_kernel<<<H, 256, 0, stream>>>(mask, first, last, clusters, coef, out);
}